// JacobiPool_37787122270444
// MI455X (gfx1250) — compile-verified
//
#include <hip/hip_runtime.h>
#include <math.h>
#include <stdint.h>

// ---------------------------------------------------------------------------
// JacobiPool for MI455X (gfx1250, wave32).
// Dense mat-vecs use V_WMMA_F32_16X16X4_F32 (exact f32 matrix path).
// Sparse passes use native f32 global atomics. Top-k via bitonic sort.
// ---------------------------------------------------------------------------

typedef __attribute__((ext_vector_type(2))) float v2f;
typedef __attribute__((ext_vector_type(8))) float v8f;

#define NEG_SLOPE 0.2f
#define EPSF 1e-16f

__device__ __forceinline__ float leaky(float v) {
    return v > 0.0f ? v : v * NEG_SLOPE;
}

// float atomic max via sign-split int trick (works with init = -inf)
__device__ __forceinline__ void atomicMaxF(float* addr, float val) {
    if (!(val < 0.0f)) {
        atomicMax((int*)addr, __float_as_int(val));
    } else {
        atomicMin((unsigned int*)addr, __float_as_uint(val));
    }
}

// ---------------------------------------------------------------------------
// init: m=-inf, s=0, agg=0, node_new=-1, zbuf=0 (zero B columns for matvec)
// ---------------------------------------------------------------------------
__global__ void jp_init_nodes(float* m, float* s, float* agg, int* node_new,
                              float* zbuf, int n, int C) {
    int i = blockIdx.x * blockDim.x + threadIdx.x;
    if (i >= n) return;
    m[i] = __int_as_float(0xFF800000);  // -inf
    s[i] = 0.0f;
    agg[i] = 0.0f;
    node_new[i] = -1;
    if (i < C) zbuf[i] = 0.0f;
}

// ---------------------------------------------------------------------------
// Dual mat-vec via WMMA f32 16x16x4:
//   xw[i]   = sum_c x[i,c] * w_att[c]
//   xhat[i] = sum_c x[i,c] * lin_w[c] + lin_b
// One wave handles 16 rows; 64 WMMA steps of K=4 cover C=256.
// A layout (16x4 f32, 2 VGPR): lanes 0-15 M=lane hold K=0,1; lanes 16-31 K=2,3.
// B layout (4x16 f32, 2 VGPR): lanes 0-15 N=lane hold K=0(b.x),1(b.y);
//                              lanes 16-31 hold K=2(b.x),3(b.y).
// D layout (16x16 f32, 8 VGPR): VGPR j, lanes 0-15 -> (M=j, N=lane),
//                               lanes 16-31 -> (M=j+8, N=lane-16).
// Column N=0 accumulates xw, N=1 accumulates xhat. All other B columns read
// from a zeroed buffer so every lane does an unconditional b64 load
// (no exec-mask churn in the hot loop).
// ---------------------------------------------------------------------------
__global__ __launch_bounds__(256) void jp_matvec_wmma(
    const float* __restrict__ x, const float* __restrict__ w_att,
    const float* __restrict__ lin_w, const float* __restrict__ lin_b,
    const float* __restrict__ zbuf,
    float* __restrict__ xw, float* __restrict__ xhat, int n, int C) {
    int gwave = (blockIdx.x * blockDim.x + threadIdx.x) >> 5;  // uniform per wave
    int lane = threadIdx.x & 31;
    int row0 = gwave * 16;
    if (row0 >= n) return;  // wave-uniform exit: EXEC stays all-1s for WMMA

    int half = lane >> 4;      // 0 -> K pair {0,1}, 1 -> K pair {2,3}
    int lrow = lane & 15;      // M for A / N for B
    int m = row0 + lrow;
    if (m >= n) m = n - 1;     // clamp: duplicate rows, never stored
    int koff = half * 2;
    const float* ap = x + (size_t)m * (size_t)C + koff;
    const float* bp = ((lrow == 0) ? w_att : (lrow == 1) ? lin_w : zbuf) + koff;

    v8f acc = {};
#pragma unroll 8
    for (int kb = 0; kb < C; kb += 4) {
        v2f a = *(const v2f*)(ap + kb);   // unconditional global_load_b64
        v2f b = *(const v2f*)(bp + kb);   // unconditional global_load_b64
        acc = __builtin_amdgcn_wmma_f32_16x16x4_f32(
            /*neg_a=*/false, a, /*neg_b=*/false, b,
            /*c_mod=*/(short)0, acc, /*reuse_a=*/false, /*reuse_b=*/false);
    }

    // lanes with N==0 write xw, N==1 write xhat (+bias); 8 rows per lane half.
    if (lrow <= 1) {
        float* dst = (lrow == 0) ? xw : xhat;
        float bias = (lrow == 0) ? 0.0f : lin_b[0];
        int rbase = row0 + half * 8;
#pragma unroll
        for (int j = 0; j < 8; ++j) {
            int r = rbase + j;
            if (r < n) dst[r] = acc[j] + bias;
        }
    }
}

// ---------------------------------------------------------------------------
// Edge pass 1: segment max of leaky-relu logits over dst.
// Edges 0..E-1 from edge_index; edges E..E+n-1 are self loops.
// ---------------------------------------------------------------------------
__global__ void jp_edge_max(const int* __restrict__ ei, const float* __restrict__ xw,
                            const float* __restrict__ att_src,
                            const float* __restrict__ att_dst,
                            float* __restrict__ m, int E, int n) {
    int e = blockIdx.x * blockDim.x + threadIdx.x;
    int total = E + n;
    if (e >= total) return;
    int s, d;
    if (e < E) { s = ei[e]; d = ei[E + e]; }
    else       { s = e - E; d = e - E; }
    float lg = leaky(att_src[0] * xw[s] + att_dst[0] * xw[d]);
    atomicMaxF(&m[d], lg);
}

// ---------------------------------------------------------------------------
// Edge pass 2: segment sum of exp(logit - m[dst]).
// ---------------------------------------------------------------------------
__global__ void jp_edge_sum(const int* __restrict__ ei, const float* __restrict__ xw,
                            const float* __restrict__ att_src,
                            const float* __restrict__ att_dst,
                            const float* __restrict__ m, float* __restrict__ ssum,
                            int E, int n) {
    int e = blockIdx.x * blockDim.x + threadIdx.x;
    int total = E + n;
    if (e >= total) return;
    int s, d;
    if (e < E) { s = ei[e]; d = ei[E + e]; }
    else       { s = e - E; d = e - E; }
    float lg = leaky(att_src[0] * xw[s] + att_dst[0] * xw[d]);
    float ev = expf(lg - m[d]);
    atomicAdd(&ssum[d], ev);
}

// ---------------------------------------------------------------------------
// Edge pass 3: Jacobi polynomial on edge values + aggregation.
// deg[n] = s[n]/(s[n]+eps) analytically (segment_sum of alpha over dst).
// Jacobi thetas for a=b=1: k=2 -> t0=1.875, t1=0, t2=0.75
//                          k=3 -> t0=56/30, t1=0, t2=0.8
// ---------------------------------------------------------------------------
__global__ void jp_edge_agg(const int* __restrict__ ei, const float* __restrict__ xw,
                            const float* __restrict__ xhat,
                            const float* __restrict__ att_src,
                            const float* __restrict__ att_dst,
                            const float* __restrict__ alphas,
                            const float* __restrict__ m, const float* __restrict__ ssum,
                            float* __restrict__ agg, int E, int n) {
    int e = blockIdx.x * blockDim.x + threadIdx.x;
    int total = E + n;
    if (e >= total) return;
    int s, d;
    float j0;
    if (e < E) { s = ei[e]; d = ei[E + e]; j0 = 0.0f; }
    else       { s = e - E; d = e - E;     j0 = 1.0f; }
    float lg = leaky(att_src[0] * xw[s] + att_dst[0] * xw[d]);
    float ev = expf(lg - m[d]);
    float sd = ssum[d];
    float ss = ssum[s];
    float alpha = ev / (sd + EPSF);
    float degd = sd / (sd + EPSF);
    float degs = ss / (ss + EPSF);
    float a = alpha * rsqrtf(degd * degs + EPSF);
    // Jacobi recursion (AP=BP=1, K=3)
    float j1 = 2.0f * a;
    float j2 = 1.875f * a * j1 - 0.75f * j0;
    float j3 = (56.0f / 30.0f) * a * j2 - 0.8f * j1;
    float poly = alphas[0] * j0 + alphas[1] * j1 + alphas[2] * j2 + alphas[3] * j3;
    atomicAdd(&agg[d], poly * xhat[s]);
}

// ---------------------------------------------------------------------------
// Pack sort keys: descending by agg value, ties by ascending node index.
// composite = (~monotonic(agg) << 32) | idx, pad with UINT64_MAX.
// ---------------------------------------------------------------------------
__global__ void jp_pack_keys(const float* __restrict__ agg,
                             unsigned long long* __restrict__ keys, int n, int total) {
    int i = blockIdx.x * blockDim.x + threadIdx.x;
    if (i >= total) return;
    if (i < n) {
        unsigned u = __float_as_uint(agg[i]);
        unsigned mk = (u & 0x80000000u) ? ~u : (u | 0x80000000u);  // ascending monotonic
        unsigned kd = ~mk;                                          // descending key
        keys[i] = ((unsigned long long)kd << 32) | (unsigned)i;
    } else {
        keys[i] = 0xFFFFFFFFFFFFFFFFULL;
    }
}

// one compare-exchange step of a global bitonic sort (ascending)
__global__ void jp_bitonic_step(unsigned long long* __restrict__ keys, int j, int k) {
    int i = blockIdx.x * blockDim.x + threadIdx.x;
    int ixj = i ^ j;
    if (ixj > i) {
        unsigned long long a = keys[i];
        unsigned long long b = keys[ixj];
        bool up = ((i & k) == 0);
        if ((a > b) == up) { keys[i] = b; keys[ixj] = a; }
    }
}

// ---------------------------------------------------------------------------
// Write perm / batch outputs and scatter node_new[perm[r]] = r.
// ---------------------------------------------------------------------------
__global__ void jp_write_perm(const unsigned long long* __restrict__ keys,
                              int* __restrict__ node_new,
                              float* __restrict__ out_perm,
                              float* __restrict__ out_batch, int kk) {
    int r = blockIdx.x * blockDim.x + threadIdx.x;
    if (r >= kk) return;
    int idx = (int)(unsigned)(keys[r] & 0xFFFFFFFFu);
    node_new[idx] = r;
    out_perm[r] = (float)idx;
    out_batch[r] = 0.0f;
}

// x_out[r, :] = x[perm[r], :] * tanh(agg[perm[r]]);  one block per row
__global__ void jp_gather_x(const float* __restrict__ x, const float* __restrict__ agg,
                            const unsigned long long* __restrict__ keys,
                            float* __restrict__ xout, int C) {
    int r = blockIdx.x;
    int c = threadIdx.x;
    int idx = (int)(unsigned)(keys[r] & 0xFFFFFFFFu);
    float t = tanhf(agg[idx]);
    xout[(size_t)r * (size_t)C + c] = x[(size_t)idx * (size_t)C + c] * t;
}

// relabel edges, emit edge_index_new rows + edge_mask (as f32)
__global__ void jp_relabel(const int* __restrict__ ei, const int* __restrict__ node_new,
                           float* __restrict__ out_src, float* __restrict__ out_dst,
                           float* __restrict__ out_mask, int E) {
    int e = blockIdx.x * blockDim.x + threadIdx.x;
    if (e >= E) return;
    int ns = node_new[ei[e]];
    int nd = node_new[ei[E + e]];
    bool keep = (ns >= 0) && (nd >= 0);
    out_src[e] = (float)(keep ? ns : -1);
    out_dst[e] = (float)(keep ? nd : -1);
    out_mask[e] = keep ? 1.0f : 0.0f;
}

static inline int cdiv(int a, int b) { return (a + b - 1) / b; }

extern "C" void kernel_launch(void* const* d_in, const int* in_sizes, int n_in,
                              void* d_out, int out_size, void* d_ws, size_t ws_size,
                              hipStream_t stream) {
    const float* x       = (const float*)d_in[0];
    const int*   ei      = (const int*)d_in[1];
    const float* w_att   = (const float*)d_in[2];
    const float* att_src = (const float*)d_in[3];
    const float* att_dst = (const float*)d_in[4];
    const float* alphas  = (const float*)d_in[5];
    const float* lin_w   = (const float*)d_in[6];
    const float* lin_b   = (const float*)d_in[7];

    const int C = in_sizes[2];            // 256
    const int N = in_sizes[0] / C;        // 100000
    const int E = in_sizes[1] / 2;        // 3200000
    const int kk = (4 * N + 4) / 5;       // ceil(0.8*N) = 80000

    // workspace layout
    float* xw       = (float*)d_ws;
    float* xhat     = xw + N;
    float* mbuf     = xhat + N;
    float* sbuf     = mbuf + N;
    float* aggbuf   = sbuf + N;
    int*   node_new = (int*)(aggbuf + N);
    float* zbuf     = (float*)(node_new + N);         // C zeroed floats (8B aligned)
    uintptr_t p = (uintptr_t)(zbuf + C);
    p = (p + 7) & ~(uintptr_t)7;
    unsigned long long* keys = (unsigned long long*)p;

    // output layout (all f32): x_out | new_src | new_dst | mask | batch | perm
    float* out_x     = (float*)d_out;
    float* out_src   = out_x + (size_t)kk * (size_t)C;
    float* out_dst   = out_src + E;
    float* out_mask  = out_dst + E;
    float* out_batch = out_mask + E;
    float* out_perm  = out_batch + kk;

    const int B = 256;
    const int ET = E + N;

    // 1. init node state (+ zero B-column buffer)
    jp_init_nodes<<<cdiv(N, B), B, 0, stream>>>(mbuf, sbuf, aggbuf, node_new, zbuf, N, C);

    // 2. dual mat-vec with WMMA (16 rows per wave, 8 waves per block)
    {
        int waves = cdiv(N, 16);
        int blocks = cdiv(waves, 8);
        jp_matvec_wmma<<<blocks, B, 0, stream>>>(x, w_att, lin_w, lin_b, zbuf,
                                                 xw, xhat, N, C);
    }

    // 3-5. edge passes (softmax max, softmax sum, jacobi aggregation)
    jp_edge_max<<<cdiv(ET, B), B, 0, stream>>>(ei, xw, att_src, att_dst, mbuf, E, N);
    jp_edge_sum<<<cdiv(ET, B), B, 0, stream>>>(ei, xw, att_src, att_dst, mbuf, sbuf, E, N);
    jp_edge_agg<<<cdiv(ET, B), B, 0, stream>>>(ei, xw, xhat, att_src, att_dst, alphas,
                                               mbuf, sbuf, aggbuf, E, N);

    // 6. top-k via bitonic sort on packed 64-bit keys
    int sortN = 1;
    while (sortN < N) sortN <<= 1;        // 131072
    jp_pack_keys<<<cdiv(sortN, B), B, 0, stream>>>(aggbuf, keys, N, sortN);
    for (int k = 2; k <= sortN; k <<= 1) {
        for (int j = k >> 1; j > 0; j >>= 1) {
            jp_bitonic_step<<<sortN / B, B, 0, stream>>>(keys, j, k);
        }
    }

    // 7. perm / batch outputs + node_new scatter
    jp_write_perm<<<cdiv(kk, B), B, 0, stream>>>(keys, node_new, out_perm, out_batch, kk);

    // 8. gated feature gather (one block per surviving row)
    jp_gather_x<<<kk, C, 0, stream>>>(x, aggbuf, keys, out_x, C);

    // 9. edge relabel + mask
    jp_relabel<<<cdiv(E, B), B, 0, stream>>>(ei, node_new, out_src, out_dst, out_mask, E);
}